// _ConvBlock4_43018392436852
// MI455X (gfx1250) — compile-verified
//
#include <hip/hip_runtime.h>

typedef __attribute__((ext_vector_type(2))) float v2f;
typedef __attribute__((ext_vector_type(8))) float v8f;

// ---------------------------------------------------------------------------
// Kernel 1: zero the output (harness poisons d_out with 0xAA before timing).
// ---------------------------------------------------------------------------
__global__ void _pool_zero_kernel(float* __restrict__ out, int n) {
    int i = blockIdx.x * blockDim.x + threadIdx.x;
    int stride = gridDim.x * blockDim.x;
    for (; i < n; i += stride) out[i] = 0.0f;
}

// ---------------------------------------------------------------------------
// Kernel 2: weighted gather + WMMA diag-scale + atomic scatter.
//
// One wave32 per group of 16 edges.
//   msgs[16x64] = diag(w[0..15]) * X_gathered[16x64]
// computed as 4 K-chunks x 4 N-tiles of V_WMMA_F32_16X16X4_F32.
//
// f32 WMMA VGPR striping (ISA 7.12.2):
//   A 16x4 : lane m (0-15): v0=K0,v1=K1 ; lane m+16: v0=K2,v1=K3   (M = lane&15)
//   B 4x16 : lanes 0-15: v0=row K0 cols 0-15, v1=row K1 ; lanes 16-31: K2/K3
//   C/D    : vgpr v: lanes 0-15 -> M=v, lanes 16-31 -> M=v+8 ; N = lane&15
// ---------------------------------------------------------------------------
__global__ __launch_bounds__(256) void _pool_scatter_kernel(
    const float* __restrict__ x,        // [N_NODES, 64]
    const int*   __restrict__ ei,       // [2, E] (src row, then dst row)
    const float* __restrict__ ea,       // [E]
    float*       __restrict__ out,      // [n_coarse, 64]
    int E, int ngroups) {

    const int lane = threadIdx.x & 31;
    const int wid  = blockIdx.x * (blockDim.x >> 5) + (threadIdx.x >> 5);
    if (wid >= ngroups) return;          // whole-wave branch: EXEC stays all-1s

    const int m  = lane & 15;            // edge-in-group / matrix M / column N
    const int hi = lane >> 4;            // lane half selects K rows {0,1} vs {2,3}

    // Per-lane edge metadata for edge (wid*16 + m); lanes 16-31 duplicate 0-15.
    int eidx = wid * 16 + m;
    const bool valid = (eidx < E);
    if (!valid) eidx = E - 1;            // branchless clamp (v_cndmask, EXEC intact)
    const int   src = ei[eidx];
    const int   dst = ei[E + eidx];
    const float w   = valid ? ea[eidx] : 0.0f;

    v8f acc[4] = {};                     // 4 N-tiles of 16 cols -> 64 features

    #pragma unroll
    for (int kc = 0; kc < 4; ++kc) {
        const int k0 = kc * 4 + hi * 2;  // global K row for v0 (v1 is k0+1)

        // A fragment: diagonal chunk, A[m,k] = w_m iff m == k (global K index)
        v2f a;
        a.x = (m == k0    ) ? w : 0.0f;
        a.y = (m == k0 + 1) ? w : 0.0f;

        // B fragment rows come from edges k0 / k0+1 of this group
        const int s0 = __shfl(src, k0,     32);
        const int s1 = __shfl(src, k0 + 1, 32);
        const float* r0 = x + (size_t)s0 * 64;
        const float* r1 = x + (size_t)s1 * 64;

        #pragma unroll
        for (int t = 0; t < 4; ++t) {
            v2f b;                       // contiguous 64B per 16 lanes -> coalesced
            b.x = r0[t * 16 + m];
            b.y = r1[t * 16 + m];
            acc[t] = __builtin_amdgcn_wmma_f32_16x16x4_f32(
                false, a, false, b, (short)0, acc[t], false, false);
        }
    }

    // D row M = v + 8*hi belongs to edge (wid*16 + M); broadcast its dst once.
    int dd[8];
    #pragma unroll
    for (int v = 0; v < 8; ++v) dd[v] = __shfl(dst, v + hi * 8, 32);

    // Scatter: 64-float output rows; lane m = column -> coalesced per dst row.
    #pragma unroll
    for (int t = 0; t < 4; ++t) {
        #pragma unroll
        for (int v = 0; v < 8; ++v) {
            atomicAdd(out + (size_t)dd[v] * 64 + t * 16 + m, acc[t][v]);
        }
    }
}

// ---------------------------------------------------------------------------
extern "C" void kernel_launch(void* const* d_in, const int* in_sizes, int n_in,
                              void* d_out, int out_size, void* d_ws, size_t ws_size,
                              hipStream_t stream) {
    const float* x   = (const float*)d_in[0];   // [100000*64] f32
    const int*   ei  = (const int*)  d_in[1];   // [2*E] int32
    const float* ea  = (const float*)d_in[2];   // [E] f32
    float*       out = (float*)d_out;           // [n_coarse*64] f32
    const int E = in_sizes[2];

    // Zero output (poisoned by harness).
    {
        int threads = 256;
        int blocks  = (out_size + threads - 1) / threads;
        if (blocks > 4096) blocks = 4096;
        _pool_zero_kernel<<<blocks, threads, 0, stream>>>(out, out_size);
    }

    // One wave per 16 edges; 8 waves (256 threads) per block.
    const int ngroups = (E + 15) / 16;
    const int blocks  = (ngroups + 7) / 8;
    _pool_scatter_kernel<<<blocks, 256, 0, stream>>>(x, ei, ea, out, E, ngroups);
}